// QuanvolutionClassifierQuantum_65481071396971
// MI455X (gfx1250) — compile-verified
//
#include <hip/hip_runtime.h>
#include <hip/hip_bf16.h>
#include <math.h>

#define IMG 224
#define STRIDE 2
#define P_SIDE 112
#define P (P_SIDE * P_SIDE)          // 12544
#define BATCH 32
#define NUM_CLASSES 10
#define N_FEAT (4 * P)               // 50176
#define KTOT N_FEAT                  // GEMM K dimension
#define K_ITERS (KTOT / 4)           // 12544 WMMA K-steps
#define GEMM_BLOCKS 448
#define ITERS_PER_BLOCK (K_ITERS / GEMM_BLOCKS)  // 28

typedef __attribute__((ext_vector_type(2))) float v2f;
typedef __attribute__((ext_vector_type(8))) float v8f;

// ---------------- 4-qubit state-vector helpers (all compile-time strides ->
// everything lives in registers, fully unrolled) --------------------------
// index = i*8 + j*4 + k*2 + l  (wire w has bit mask 8>>w)

template <int ST>
__device__ __forceinline__ void ry_g(float* re, float* im, float c, float s) {
#pragma unroll
  for (int i = 0; i < 16; ++i) {
    if ((i & ST) == 0) {
      const int j = i + ST;
      float r0 = re[i], i0 = im[i], r1 = re[j], i1 = im[j];
      re[i] = c * r0 - s * r1;  im[i] = c * i0 - s * i1;
      re[j] = s * r0 + c * r1;  im[j] = s * i0 + c * i1;
    }
  }
}

template <int ST>
__device__ __forceinline__ void rx_g(float* re, float* im, float c, float s) {
#pragma unroll
  for (int i = 0; i < 16; ++i) {
    if ((i & ST) == 0) {
      const int j = i + ST;
      float r0 = re[i], i0 = im[i], r1 = re[j], i1 = im[j];
      // a0' = c*a0 - i*s*a1 ; a1' = -i*s*a0 + c*a1
      re[i] = c * r0 + s * i1;  im[i] = c * i0 - s * r1;
      re[j] = s * i0 + c * r1;  im[j] = -s * r0 + c * i1;
    }
  }
}

template <int ST>
__device__ __forceinline__ void rz_g(float* re, float* im, float c, float s) {
#pragma unroll
  for (int i = 0; i < 16; ++i) {
    if ((i & ST) == 0) {
      const int j = i + ST;
      float r0 = re[i], i0 = im[i], r1 = re[j], i1 = im[j];
      // a0 *= (c - i s) ; a1 *= (c + i s)
      re[i] = c * r0 + s * i0;  im[i] = c * i0 - s * r0;
      re[j] = c * r1 - s * i1;  im[j] = c * i1 + s * r1;
    }
  }
}

template <int SC, int ST>
__device__ __forceinline__ void cnot_g(float* re, float* im) {
#pragma unroll
  for (int i = 0; i < 16; ++i) {
    if ((i & SC) != 0 && (i & ST) == 0) {
      const int j = i + ST;
      float t = re[i]; re[i] = re[j]; re[j] = t;
      t = im[i]; im[i] = im[j]; im[j] = t;
    }
  }
}

// cs[12] = {cos(p0/2), sin(p0/2), cos(p1/2), sin(p1/2), ...}
__device__ __forceinline__ void quantum_block(const float a[4], const float* cs,
                                              float out[4]) {
  float cw[4], sw[4];
#pragma unroll
  for (int w = 0; w < 4; ++w) sincosf(0.5f * a[w], &sw[w], &cw[w]);

  float re[16], im[16];
#pragma unroll
  for (int i = 0; i < 16; ++i) {
    float v = ((i & 8) ? sw[0] : cw[0]) * ((i & 4) ? sw[1] : cw[1]) *
              ((i & 2) ? sw[2] : cw[2]) * ((i & 1) ? sw[3] : cw[3]);
    re[i] = v;
    im[i] = 0.0f;
  }

  ry_g<8>(re, im, cs[0], cs[1]);    // RY(p0) wire 0
  rx_g<4>(re, im, cs[2], cs[3]);    // RX(p1) wire 1
  cnot_g<8, 4>(re, im);             // CNOT(0,1)
  rz_g<2>(re, im, cs[4], cs[5]);    // RZ(p2) wire 2
  ry_g<1>(re, im, cs[6], cs[7]);    // RY(p3) wire 3
  cnot_g<2, 1>(re, im);             // CNOT(2,3)
  rx_g<2>(re, im, cs[8], cs[9]);    // RX(p4) wire 2
  rz_g<8>(re, im, cs[10], cs[11]);  // RZ(p5) wire 0

  float m0 = 0.f, m1 = 0.f, m2 = 0.f, m3 = 0.f;
#pragma unroll
  for (int i = 0; i < 16; ++i) {
    float pr = re[i] * re[i] + im[i] * im[i];
    m0 += (i & 8) ? -pr : pr;
    m1 += (i & 4) ? -pr : pr;
    m2 += (i & 2) ? -pr : pr;
    m3 += (i & 1) ? -pr : pr;
  }
  out[0] = m0; out[1] = m1; out[2] = m2; out[3] = m3;
}

// ---------------- Kernel 1: per-patch quantum feature extraction ----------
__global__ void quanv_kernel(const float* __restrict__ x,
                             const float* __restrict__ p1,
                             const float* __restrict__ p2,
                             float* __restrict__ feats) {
  int tid = blockIdx.x * blockDim.x + threadIdx.x;
  if (tid >= BATCH * P) return;
  int b = tid / P;
  int p = tid - b * P;
  int pr = p / P_SIDE;
  int pc = p - pr * P_SIDE;
  int R = pr * STRIDE;
  int C = pc * STRIDE;
  const float* xb = x + b * IMG * IMG;

  float f[4];
  f[0] = xb[R * IMG + C];
  f[1] = xb[R * IMG + C + 1];
  bool edge = (C == IMG - STRIDE);
  f[2] = edge ? xb[(R + 1) * IMG + C] : xb[R * IMG + C + 2];
  f[3] = edge ? xb[(R + 1) * IMG + C + 1] : xb[(R + 1) * IMG + C];

  float cs1[12], cs2[12];
#pragma unroll
  for (int i = 0; i < 6; ++i) {
    sincosf(0.5f * p1[i], &cs1[2 * i + 1], &cs1[2 * i]);  // [cos, sin]
    sincosf(0.5f * p2[i], &cs2[2 * i + 1], &cs2[2 * i]);
  }

  float m[4], m2v[4];
  quantum_block(f, cs1, m);
  quantum_block(m, cs2, m2v);

  float* fr = feats + (size_t)b * N_FEAT + 4 * p;
  fr[0] = m[0] + m2v[0];
  fr[1] = m[1] + m2v[1];
  fr[2] = m[2] + m2v[2];
  fr[3] = m[3] + m2v[3];
}

// ---------------- Kernel 2: K-split f32 WMMA GEMM (32x16 out, K=50176) ----
// A fragment (16x4 f32): lanes 0-15 hold {K=0,K=1} for M=lane,
//                        lanes 16-31 hold {K=2,K=3} for M=lane-16.
// B fragment (4x16 f32): lanes 0-15 hold {K=0,K=1} for N=lane, etc.
// C/D (16x16 f32): VGPR r, lane<16 -> (M=r, N=lane); lane>=16 -> (M=r+8, N=lane-16).
//
// N-padding (classes 10..15) is done branchlessly: every lane loads from a
// clamped-valid W row and scales the fragment by 0/1, so no exec-mask
// diamonds appear in the hot loop and EXEC stays all-ones for WMMA.
__global__ void __launch_bounds__(32) gemm_wmma_kernel(
    const float* __restrict__ feats, const float* __restrict__ W,
    float* __restrict__ acc) {
  const int lane = threadIdx.x;
  const int n = lane & 15;
  const int khalf = (lane >> 4);        // 0 -> K{0,1}, 1 -> K{2,3}
  const int koff = khalf * 2;
  const int iter0 = blockIdx.x * ITERS_PER_BLOCK;

  const int nc = (n < NUM_CLASSES) ? n : 0;        // clamped valid row
  const float bscale = (n < NUM_CLASSES) ? 1.0f : 0.0f;

  const float* arow0 = feats + (size_t)n * KTOT + koff;
  const float* arow1 = feats + (size_t)(n + 16) * KTOT + koff;
  const float* brow  = W + (size_t)nc * KTOT + koff;

  v8f c0 = {};
  v8f c1 = {};

  for (int it = 0; it < ITERS_PER_BLOCK; ++it) {
    const int k = (iter0 + it) * 4;

    v2f a0 = *(const v2f*)(arow0 + k);   // rows 0..15
    v2f a1 = *(const v2f*)(arow1 + k);   // rows 16..31
    v2f bm = *(const v2f*)(brow + k);
    bm.x *= bscale;
    bm.y *= bscale;

    c0 = __builtin_amdgcn_wmma_f32_16x16x4_f32(false, a0, false, bm,
                                               (short)0, c0, false, false);
    c1 = __builtin_amdgcn_wmma_f32_16x16x4_f32(false, a1, false, bm,
                                               (short)0, c1, false, false);
  }

  const int mh = khalf * 8;
#pragma unroll
  for (int r = 0; r < 8; ++r) {
    atomicAdd(&acc[(r + mh) * 16 + n], c0[r]);
    atomicAdd(&acc[(16 + r + mh) * 16 + n], c1[r]);
  }
}

// ---------------- Kernel 0: zero the accumulator --------------------------
__global__ void init_acc_kernel(float* __restrict__ acc) {
  int i = blockIdx.x * blockDim.x + threadIdx.x;
  if (i < BATCH * 16) acc[i] = 0.0f;
}

// ---------------- Kernel 3: bias + log_softmax ----------------------------
__global__ void finalize_kernel(const float* __restrict__ acc,
                                const float* __restrict__ bias,
                                float* __restrict__ out) {
  int b = threadIdx.x;
  if (b >= BATCH) return;
  float lg[NUM_CLASSES];
  float mx = -INFINITY;
#pragma unroll
  for (int c = 0; c < NUM_CLASSES; ++c) {
    lg[c] = acc[b * 16 + c] + bias[c];
    mx = fmaxf(mx, lg[c]);
  }
  float se = 0.0f;
#pragma unroll
  for (int c = 0; c < NUM_CLASSES; ++c) se += expf(lg[c] - mx);
  float lse = mx + logf(se);
#pragma unroll
  for (int c = 0; c < NUM_CLASSES; ++c) out[b * NUM_CLASSES + c] = lg[c] - lse;
}

extern "C" void kernel_launch(void* const* d_in, const int* in_sizes, int n_in,
                              void* d_out, int out_size, void* d_ws,
                              size_t ws_size, hipStream_t stream) {
  (void)in_sizes; (void)n_in; (void)out_size; (void)ws_size;
  const float* x  = (const float*)d_in[0];   // (32, 224, 224)
  const float* p1 = (const float*)d_in[1];   // (6,)
  const float* p2 = (const float*)d_in[2];   // (6,)
  const float* W  = (const float*)d_in[3];   // (10, 50176)
  const float* bi = (const float*)d_in[4];   // (10,)
  float* out = (float*)d_out;                // (32, 10)

  // workspace layout: feats [32 * 50176] floats, then acc [32*16] floats
  float* feats = (float*)d_ws;
  float* acc = feats + (size_t)BATCH * N_FEAT;

  init_acc_kernel<<<1, 512, 0, stream>>>(acc);

  int total = BATCH * P;
  quanv_kernel<<<(total + 255) / 256, 256, 0, stream>>>(x, p1, p2, feats);

  gemm_wmma_kernel<<<GEMM_BLOCKS, 32, 0, stream>>>(feats, W, acc);

  finalize_kernel<<<1, 32, 0, stream>>>(acc, bi, out);
}